// InnerConvexViolationProjection_73203422593449
// MI455X (gfx1250) — compile-verified
//
#include <hip/hip_runtime.h>
#include <hip/hip_bf16.h>
#include <cstdint>

// ---------------------------------------------------------------------------
// InnerConvexViolationProjection on MI455X (gfx1250)
//   grid  = 2048 blocks (one per (b,s) problem), block = 256 threads (8 waves)
//   - TDM tensor_load_to_lds stages the 128KB A tile (pad feature -> stride 260)
//   - Gram matrix G = A*A^T built with V_WMMA_F32_16X16X4_F32, kept in VGPRs
//   - K-loop iterates in residual space: r <- r - eta*G*relu(r)  (exact algebra)
// ---------------------------------------------------------------------------

#define MDIM 128
#define NDIM 256
#define AS   260          // padded LDS row stride (floats): banks spread 4i mod 64
#define KIT  30
#define PIT  5
#define NTHR 256

typedef float        v8f __attribute__((ext_vector_type(8)));
typedef float        v2f __attribute__((ext_vector_type(2)));
typedef unsigned int v4u __attribute__((ext_vector_type(4)));
typedef int          v4i __attribute__((ext_vector_type(4)));
typedef int          v8i __attribute__((ext_vector_type(8)));

__device__ __forceinline__ float bsum(float v, float* s8, int t) {
#pragma unroll
  for (int o = 16; o > 0; o >>= 1) v += __shfl_down(v, o, 32);
  __syncthreads();
  if ((t & 31) == 0) s8[t >> 5] = v;
  __syncthreads();
  float r = 0.f;
#pragma unroll
  for (int w = 0; w < 8; ++w) r += s8[w];
  return r;
}

__device__ __forceinline__ float bmax(float v, float* s8, int t) {
#pragma unroll
  for (int o = 16; o > 0; o >>= 1) v = fmaxf(v, __shfl_down(v, o, 32));
  __syncthreads();
  if ((t & 31) == 0) s8[t >> 5] = v;
  __syncthreads();
  float r = s8[0];
#pragma unroll
  for (int w = 1; w < 8; ++w) r = fmaxf(r, s8[w]);
  return r;
}

__device__ __forceinline__ float bmin(float v, float* s8, int t) {
#pragma unroll
  for (int o = 16; o > 0; o >>= 1) v = fminf(v, __shfl_down(v, o, 32));
  __syncthreads();
  if ((t & 31) == 0) s8[t >> 5] = v;
  __syncthreads();
  float r = s8[0];
#pragma unroll
  for (int w = 1; w < 8; ++w) r = fminf(r, s8[w]);
  return r;
}

// y[i] = row_i(sA) . vec  -> srr[0..127]; 2 threads per row (halves via sp)
__device__ __forceinline__ void matvec_row(const float* sA, const float* vec,
                                           float* sp, float* srr, int t) {
  __syncthreads();
  const int i = t & 127, h = t >> 7;
  const float* row = &sA[i * AS + h * 128];
  const float* v   = vec + h * 128;
  float s = 0.f;
#pragma unroll 4
  for (int c = 0; c < 128; c += 4) {
    float4 a4 = *(const float4*)(row + c);
    float4 v4 = *(const float4*)(v + c);
    s = fmaf(a4.x, v4.x, s); s = fmaf(a4.y, v4.y, s);
    s = fmaf(a4.z, v4.z, s); s = fmaf(a4.w, v4.w, s);
  }
  sp[t] = s;
  __syncthreads();
  if (t < 128) srr[t] = sp[t] + sp[t + 128];
  __syncthreads();
}

// g[t] = col_t(sA) . r   (thread t owns column t; lanes hit distinct banks)
__device__ __forceinline__ float matvec_col(const float* sA, const float* r, int t) {
  float s = 0.f;
#pragma unroll 8
  for (int i = 0; i < MDIM; ++i) s = fmaf(sA[i * AS + t], r[i], s);
  return s;
}

__global__ void __launch_bounds__(NTHR)
InnerConvexViolationProjection_kernel(const float* __restrict__ X,
                                      const float* __restrict__ A,
                                      const float* __restrict__ Bv,
                                      float* __restrict__ outX,
                                      float* __restrict__ outM) {
  __shared__ float sA[MDIM * AS];   // 130 KB: row-normalized A, padded rows
  __shared__ float sx[NDIM];
  __shared__ float sd[NDIM];
  __shared__ float sp[NDIM];
  __shared__ float sp2[NDIM];
  __shared__ float srr[MDIM];
  __shared__ float sbw[MDIM];
  __shared__ float sred[8];

  const int t = threadIdx.x;
  const int p = blockIdx.x;
  const uint64_t aBase = (uint64_t)p * (uint64_t)(MDIM * NDIM);

  // ---- Stage A -> LDS via Tensor Data Mover (pad 4 dwords per 256-dword row)
#if __has_builtin(__builtin_amdgcn_tensor_load_to_lds)
  if (t < 32) {  // one wave issues the DMA (TDM ignores EXEC; issue once)
    const uint64_t ga  = (uint64_t)(uintptr_t)(A + aBase);
    const uint32_t lds = (uint32_t)(uintptr_t)(const void*)&sA[0];
    v4u g0;
    g0[0] = 1u;                                         // count=1
    g0[1] = lds;                                        // lds_addr
    g0[2] = (uint32_t)ga;                               // global_addr[31:0]
    g0[3] = (uint32_t)((ga >> 32) & 0x01FFFFFFu) | (2u << 30); // addr hi | type=2
    v8i g1;
    g1[0] = (int)((2u << 16) | (1u << 20) | (7u << 22) | (3u << 25));
    //         data_size=4B | pad_en | pad_interval=256dw | pad_amount=4dw
    g1[1] = (int)(256u << 16);   // tensor_dim0 = 256
    g1[2] = (int)(128u << 16);   // tensor_dim1 = 128
    g1[3] = (int)(256u << 16);   // tile_dim0   = 256
    g1[4] = (int)128u;           // tile_dim1   = 128
    g1[5] = (int)256u;           // tensor_dim0_stride = 256
    g1[6] = 0; g1[7] = 0;
    v4i z4 = {0, 0, 0, 0};
#if __clang_major__ >= 23
    v8i z8 = {0, 0, 0, 0, 0, 0, 0, 0};
    __builtin_amdgcn_tensor_load_to_lds(g0, g1, z4, z4, z8, 0);
#else
    __builtin_amdgcn_tensor_load_to_lds(g0, g1, z4, z4, 0);
#endif
#if __has_builtin(__builtin_amdgcn_s_wait_tensorcnt)
    __builtin_amdgcn_s_wait_tensorcnt(0);
#else
    asm volatile("s_wait_tensorcnt 0x0" ::: "memory");
#endif
  }
#else
  for (int idx = t; idx < MDIM * (NDIM / 4); idx += NTHR) {
    const int r = idx >> 6, c = idx & 63;  // 64 float4 per row
    float4 val = ((const float4*)(A + aBase + (size_t)r * NDIM))[c];
    *(float4*)&sA[r * AS + 4 * c] = val;
  }
#endif
  __syncthreads();

  // ---- Row-normalize: A_w = A/||row||, b_w = b/||row||  (clip 1e-12)
  {
    const int i = t & 127, h = t >> 7;
    float* row = &sA[i * AS + h * 128];
    float s = 0.f;
#pragma unroll 4
    for (int c = 0; c < 128; c += 4) {
      float4 a4 = *(const float4*)(row + c);
      s = fmaf(a4.x, a4.x, s); s = fmaf(a4.y, a4.y, s);
      s = fmaf(a4.z, a4.z, s); s = fmaf(a4.w, a4.w, s);
    }
    sp[t] = s;
    __syncthreads();
    if (t < 128) {
      float nrm = fmaxf(sqrtf(sp[t] + sp[t + 128]), 1e-12f);
      float inv = 1.0f / nrm;
      srr[t] = inv;                                 // stash inv-norm
      sbw[t] = Bv[(size_t)p * MDIM + t] * inv;      // b_w
    }
    __syncthreads();
    const float inv = srr[i];
#pragma unroll 4
    for (int c = 0; c < 128; c += 4) {
      float4 a4 = *(const float4*)(row + c);
      a4.x *= inv; a4.y *= inv; a4.z *= inv; a4.w *= inv;
      *(float4*)(row + c) = a4;
    }
    __syncthreads();
  }

  const float x0 = X[(size_t)p * NDIM + t];   // thread t owns x-element t

  // ---- Power iteration for eta = 0.99 / (||A v||^2 + 1e-12)
  sx[t] = 0.0625f;  // 1/sqrt(256)
  for (int it = 0; it < PIT; ++it) {
    matvec_row(sA, sx, sp, srr, t);                 // srr = A v
    float atav = matvec_col(sA, srr, t);            // (A^T A v)[t]
    float nn = bsum(atav * atav, sred, t);
    sx[t] = atav / (sqrtf(nn) + 1e-12f);
  }
  matvec_row(sA, sx, sp, srr, t);                   // srr = A v (final)
  float sq = 0.f;
  if (t < 128) { const float av = srr[t]; sq = av * av; }
  const float num = bsum(sq, sred, t);
  const float eta = 0.99f / (num + 1e-12f);

  // ---- Gram matrix G = A_w A_w^T via WMMA f32 16x16x4; C-fragments stay in VGPRs
  const int lane = t & 31, wave = t >> 5;
  const int fm = lane & 15, fhi = lane >> 4;
  v8f acc[8];
  {
    v8f z;
#pragma unroll
    for (int q = 0; q < 8; ++q) z[q] = 0.f;
#pragma unroll
    for (int J = 0; J < 8; ++J) acc[J] = z;
  }
#if __has_builtin(__builtin_amdgcn_wmma_f32_16x16x4_f32)
  {
    const int arow = (wave * 16 + fm) * AS;         // A-frag: m=lane&15, k=v+2*hi
    for (int k0 = 0; k0 < NDIM; k0 += 4) {
      v2f af = *(const v2f*)&sA[arow + k0 + 2 * fhi];
#pragma unroll
      for (int J = 0; J < 8; ++J) {                 // B-frag: n=lane&15, k=v+2*hi
        v2f bf = *(const v2f*)&sA[(J * 16 + fm) * AS + k0 + 2 * fhi];
        acc[J] = __builtin_amdgcn_wmma_f32_16x16x4_f32(
            false, af, false, bf, (short)0, acc[J], false, false);
      }
    }
  }
#else
  {
#pragma unroll
    for (int J = 0; J < 8; ++J) {
#pragma unroll
      for (int v = 0; v < 8; ++v) {
        const float* ra = &sA[(wave * 16 + v + 8 * fhi) * AS];
        const float* rb = &sA[(J * 16 + fm) * AS];
        float s = 0.f;
        for (int k = 0; k < NDIM; ++k) s = fmaf(ra[k], rb[k], s);
        acc[J][v] = s;                              // C layout: m=v+8*hi, n=lane&15
      }
    }
  }
#endif

  // ---- r0 = A_w x0 - (b_w - mu)
  sx[t] = x0;
  matvec_row(sA, sx, sp, srr, t);
  float r_i = 0.f, sumrelu = 0.f;
  if (t < 128) r_i = srr[t] - (sbw[t] - 1e-3f);

  // ---- K-loop in residual space: r <- r - eta * G * relu(r)
  float* outMp = outM + (size_t)p * (KIT * MDIM);
  for (int k = 0; k < KIT; ++k) {
    __syncthreads();
    if (t < 128) {
      outMp[(size_t)k * MDIM + t] = (r_i > 0.f) ? 1.f : 0.f;  // mask
      const float rr = fmaxf(r_i, 0.f);
      sumrelu += rr;
      srr[t] = rr;
    }
    __syncthreads();
    float y[8];
#pragma unroll
    for (int v = 0; v < 8; ++v) y[v] = 0.f;
#pragma unroll
    for (int J = 0; J < 8; ++J) {
      const float rrv = srr[J * 16 + fm];
#pragma unroll
      for (int v = 0; v < 8; ++v) y[v] = fmaf(acc[J][v], rrv, y[v]);
    }
#pragma unroll
    for (int off = 1; off < 16; off <<= 1) {
#pragma unroll
      for (int v = 0; v < 8; ++v) y[v] += __shfl_xor(y[v], off, 32);
    }
    if (fm == 0) {
      const int rowbase = wave * 16 + 8 * fhi;
#pragma unroll
      for (int v = 0; v < 8; ++v) sp[rowbase + v] = y[v];
    }
    __syncthreads();
    if (t < 128) r_i -= eta * sp[t];
  }

  // ---- x_f = x0 - eta * A_w^T (sum relu r)
  __syncthreads();
  if (t < 128) srr[t] = sumrelu;
  __syncthreads();
  const float gsum = matvec_col(sA, srr, t);
  const float xf = x0 - eta * gsum;

  // ---- Alpha-map post-processing (dual matvec: A x_f and A d in one pass)
  __syncthreads();
  sx[t] = xf;
  sd[t] = x0 - xf;
  __syncthreads();
  {
    const int i = t & 127, h = t >> 7;
    const float* row = &sA[i * AS + h * 128];
    const float* xv = &sx[h * 128];
    const float* dv = &sd[h * 128];
    float s1 = 0.f, s2 = 0.f;
#pragma unroll 4
    for (int c = 0; c < 128; c += 4) {
      float4 a4 = *(const float4*)(row + c);
      float4 x4 = *(const float4*)(xv + c);
      float4 d4 = *(const float4*)(dv + c);
      s1 = fmaf(a4.x, x4.x, s1); s1 = fmaf(a4.y, x4.y, s1);
      s1 = fmaf(a4.z, x4.z, s1); s1 = fmaf(a4.w, x4.w, s1);
      s2 = fmaf(a4.x, d4.x, s2); s2 = fmaf(a4.y, d4.y, s2);
      s2 = fmaf(a4.z, d4.z, s2); s2 = fmaf(a4.w, d4.w, s2);
    }
    sp[t] = s1; sp2[t] = s2;
  }
  __syncthreads();
  float viol = 0.f, alph = __builtin_inff();
  if (t < 128) {
    const float Ax = sp[t] + sp[t + 128];
    const float Ad = sp2[t] + sp2[t + 128];
    const float bw = sbw[t];
    viol = fmaxf(Ax - bw, 0.f);
    alph = (Ad > 0.f) ? ((bw - Ax) / (Ad + 1e-12f)) : __builtin_inff();
  }
  const float mv = bmax(viol, sred, t);
  const float am = bmin(alph, sred, t);
  float alpha = __builtin_isfinite(am) ? am : 1.0f;
  alpha = fminf(fmaxf(alpha - 1e-6f, 0.f), 1.f);
  const float xo = (mv <= 1e-7f) ? (xf + alpha * (x0 - xf)) : xf;
  outX[(size_t)p * NDIM + t] = xo;
}

extern "C" void kernel_launch(void* const* d_in, const int* in_sizes, int n_in,
                              void* d_out, int out_size, void* d_ws, size_t ws_size,
                              hipStream_t stream) {
  (void)n_in; (void)out_size; (void)d_ws; (void)ws_size;
  const float* X  = (const float*)d_in[0];
  const float* A  = (const float*)d_in[1];
  const float* Bv = (const float*)d_in[2];
  const int P = in_sizes[0] / NDIM;       // 2048 problems
  float* outX = (float*)d_out;
  float* outM = outX + (size_t)P * NDIM;  // masks follow x_out, flat (B,S,K,m)
  InnerConvexViolationProjection_kernel<<<dim3(P), dim3(NTHR), 0, stream>>>(
      X, A, Bv, outX, outM);
}